// BoxFilter_11012296146962
// MI455X (gfx1250) — compile-verified
//
#include <hip/hip_runtime.h>

// ---------- builtin probes (results show up as warnings in stderr) ----------
#if !__has_builtin(__builtin_amdgcn_wmma_f32_16x16x4_f32)
#warning "PROBE: missing __builtin_amdgcn_wmma_f32_16x16x4_f32"
#endif
#if !__has_builtin(__builtin_amdgcn_global_load_async_to_lds_b128)
#warning "PROBE: missing __builtin_amdgcn_global_load_async_to_lds_b128"
#endif
#if !__has_builtin(__builtin_amdgcn_s_wait_asynccnt)
#warning "PROBE: missing __builtin_amdgcn_s_wait_asynccnt"
#endif
#if !__has_builtin(__builtin_amdgcn_tensor_load_to_lds)
#warning "PROBE: missing __builtin_amdgcn_tensor_load_to_lds"
#endif

#if __has_builtin(__builtin_amdgcn_wmma_f32_16x16x4_f32)
#define HAVE_WMMA4 1
#else
#define HAVE_WMMA4 0
#endif

typedef __attribute__((ext_vector_type(2))) float v2f;
typedef __attribute__((ext_vector_type(4))) float v4f;
typedef __attribute__((ext_vector_type(8))) float v8f;

#define RAD 4
#define TH 32
#define TW 128
#define IN_ROWS (TH + 2 * RAD)    // 40 input rows (with vertical halo)
#define IN_COLS 144               // cols [w0-8, w0+136): 16B-aligned halo
#define IN_CHUNKS (IN_COLS / 4)   // 36 x b128 chunks per row
#define IN_STRIDE 148             // floats; 592B row pitch (16B multiple)
#define RS_STRIDE 132             // rowsum pitch (banks decorrelated)

__launch_bounds__(256)
__global__ void box_sum_9x9(const float* __restrict__ x,
                            float* __restrict__ out,
                            int H, int W) {
  __shared__ __align__(16) float s_in[IN_ROWS * IN_STRIDE];
  __shared__ float s_rs[IN_ROWS * RS_STRIDE];

  const int tid  = threadIdx.x;
  const int lane = tid & 31;
  const int wave = tid >> 5;             // 0..7 -> 16-wide column block
  const int w0 = blockIdx.x * TW;
  const int h0 = blockIdx.y * TH;
  const long long plane = (long long)blockIdx.z * H * W;
  const float* __restrict__ xp = x + plane;
  float* __restrict__ op = out + plane;

  // ---- Phase 1: global -> LDS via CDNA5 async copies; zero-fill OOB halo ----
  for (int idx = tid; idx < IN_ROWS * IN_CHUNKS; idx += 256) {
    const int row = idx / IN_CHUNKS;
    const int ch  = idx - row * IN_CHUNKS;
    const int gr  = h0 - RAD + row;
    const int gc  = w0 - 8 + ch * 4;     // chunks are fully in- or out-of-range
    float* lp = &s_in[row * IN_STRIDE + ch * 4];
    if ((unsigned)gr < (unsigned)H && (unsigned)gc < (unsigned)W) {
      unsigned lds_addr = (unsigned)(unsigned long long)lp;  // low 32b = LDS offset
      unsigned long long gaddr =
          (unsigned long long)(xp + (long long)gr * W + gc);
      asm volatile("global_load_async_to_lds_b128 %0, %1, off"
                   :: "v"(lds_addr), "v"(gaddr) : "memory");
    } else {
      v4f z = {0.f, 0.f, 0.f, 0.f};
      *(v4f*)lp = z;                     // truncated-window semantics
    }
  }
  asm volatile("s_wait_asynccnt 0" ::: "memory");
  __syncthreads();

  // ---- Phase 2: horizontal 9-tap sums, sliding window (2 LDS reads/output) --
  for (int idx = tid; idx < IN_ROWS * 8; idx += 256) {
    const int r  = idx >> 3;
    const int c0 = (idx & 7) << 4;       // 16-output run per task
    const float* rp = &s_in[r * IN_STRIDE];
    float s = 0.f;
#pragma unroll
    for (int j = 0; j < 9; ++j) s += rp[c0 + 4 + j];
    float* rsp = &s_rs[r * RS_STRIDE + c0];
    rsp[0] = s;
#pragma unroll
    for (int i = 1; i < 16; ++i) {
      s += rp[c0 + 12 + i] - rp[c0 + 3 + i];
      rsp[i] = s;
    }
  }
  __syncthreads();

  // ---- Phase 3: vertical 9-tap sums. Out(16x16) = BandS(16x24) x RS(24x16) --
  const int n    = lane & 15;            // output column within 16-wide block
  const int half = lane >> 4;            // 0 or 1 (lane-half)
  const int colg = w0 + wave * 16 + n;   // always < W (W % 128 == 0)
  const int cb   = wave * 16 + n;        // rowsum column

#if HAVE_WMMA4
  v8f acc0 = {0, 0, 0, 0, 0, 0, 0, 0};   // tile rows h0 .. h0+15
  v8f acc1 = {0, 0, 0, 0, 0, 0, 0, 0};   // tile rows h0+16 .. h0+31
#pragma unroll
  for (int k0 = 0; k0 < 6; ++k0) {       // accumulate K = 24 in chunks of 4
    const int kx = 4 * k0 + 2 * half;    // K held in .x (A VGPR0 layout)
    const int ky = kx + 1;               // K held in .y (A VGPR1 layout)
    const int m  = n;                    // A-matrix M = lane & 15
    v2f a;                               // band matrix: S[m][k] = (m<=k<=m+8)
    a.x = (kx >= m && kx <= m + 8) ? 1.0f : 0.0f;
    a.y = (ky >= m && ky <= m + 8) ? 1.0f : 0.0f;
    v2f b0, b1;                          // B 4x16: VGPR0 = K{0,2}, VGPR1 = K{1,3}
    b0.x = s_rs[kx * RS_STRIDE + cb];
    b0.y = s_rs[ky * RS_STRIDE + cb];
    b1.x = s_rs[(16 + kx) * RS_STRIDE + cb];
    b1.y = s_rs[(16 + ky) * RS_STRIDE + cb];
    acc0 = __builtin_amdgcn_wmma_f32_16x16x4_f32(
        false, a, false, b0, (short)0, acc0, false, false);
    acc1 = __builtin_amdgcn_wmma_f32_16x16x4_f32(
        false, a, false, b1, (short)0, acc1, false, false);
  }
  // D layout: VGPR vg -> M = vg + 8*half, N = lane & 15
#pragma unroll
  for (int vg = 0; vg < 8; ++vg) {
    const int m = vg + half * 8;
    int h = h0 + m;
    if (h < H) op[(long long)h * W + colg] = acc0[vg];
    h = h0 + 16 + m;
    if (h < H) op[(long long)h * W + colg] = acc1[vg];
  }
#else
  // VALU fallback: sliding window down the column (same work split as WMMA path)
#pragma unroll
  for (int tr = 0; tr < 2; ++tr) {
    const int rb = tr * 16 + half * 8;   // first rowsum row for this lane
    float s = 0.f;
#pragma unroll
    for (int k = 0; k < 9; ++k) s += s_rs[(rb + k) * RS_STRIDE + cb];
    int h = h0 + rb;
    if (h < H) op[(long long)h * W + colg] = s;
#pragma unroll
    for (int m = 1; m < 8; ++m) {
      s += s_rs[(rb + 8 + m) * RS_STRIDE + cb]
         - s_rs[(rb + m - 1) * RS_STRIDE + cb];
      h = h0 + rb + m;
      if (h < H) op[(long long)h * W + colg] = s;
    }
  }
#endif
}

extern "C" void kernel_launch(void* const* d_in, const int* in_sizes, int n_in,
                              void* d_out, int out_size, void* d_ws, size_t ws_size,
                              hipStream_t stream) {
  (void)d_ws; (void)ws_size; (void)n_in; (void)out_size;
  const float* x = (const float*)d_in[0];   // (B,C,H,W) fp32
  // d_in[1] is r (=4), baked in as RAD at compile time.
  float* out = (float*)d_out;
  const int H = 1080, W = 1920;
  const int planes = in_sizes[0] / (H * W); // B*C = 24
  dim3 grid((W + TW - 1) / TW, (H + TH - 1) / TH, planes); // 15 x 34 x 24
  box_sum_9x9<<<grid, 256, 0, stream>>>(x, out, H, W);
}